// RouterTopK_17532056502441
// MI455X (gfx1250) — compile-verified
//
#include <hip/hip_runtime.h>
#include <hip/hip_bf16.h>

typedef __attribute__((ext_vector_type(2))) float v2f;
typedef __attribute__((ext_vector_type(8))) float v8f;
typedef __attribute__((ext_vector_type(4))) int   v4i;

#define T_TOKENS 16384   // S*B = 4096*4
#define H_DIM    2048
#define E_EXP    64
#define KB       32      // K tile staged in LDS per iteration
#define BLOCK_M  64      // tokens per block (4 waves x 16 rows)
#define LDS_STR  36      // padded row stride in floats: 144B, 16B aligned, bank-friendly

#if defined(__has_builtin)
#  if __has_builtin(__builtin_amdgcn_global_load_async_to_lds_b128)
#    define HAVE_ASYNC_LDS 1
#  endif
#endif
#ifndef HAVE_ASYNC_LDS
#  define HAVE_ASYNC_LDS 0
#endif

__device__ __forceinline__ void wait_async0() {
#if __has_builtin(__builtin_amdgcn_s_wait_asynccnt)
    __builtin_amdgcn_s_wait_asynccnt(0);
#else
    asm volatile("s_wait_asynccnt 0x0" ::: "memory");
#endif
}

// 8 K-steps of 4 WMMAs from one staged K-tile; A fragment reused across 4 N tiles.
__device__ __forceinline__ void mma_tile(const float* sA, const float* sB,
                                         int wave, int lhalf, int khi,
                                         v8f& acc0, v8f& acc1, v8f& acc2, v8f& acc3)
{
    #pragma unroll
    for (int kk = 0; kk < KB; kk += 4) {
        const v2f a  = *(const v2f*)(sA + (wave * 16 + lhalf) * LDS_STR + kk + khi);
        const v2f b0 = *(const v2f*)(sB + ( 0 + lhalf) * LDS_STR + kk + khi);
        const v2f b1 = *(const v2f*)(sB + (16 + lhalf) * LDS_STR + kk + khi);
        const v2f b2 = *(const v2f*)(sB + (32 + lhalf) * LDS_STR + kk + khi);
        const v2f b3 = *(const v2f*)(sB + (48 + lhalf) * LDS_STR + kk + khi);
        acc0 = __builtin_amdgcn_wmma_f32_16x16x4_f32(false, a, false, b0, (short)0, acc0, false, false);
        acc1 = __builtin_amdgcn_wmma_f32_16x16x4_f32(false, a, false, b1, (short)0, acc1, false, false);
        acc2 = __builtin_amdgcn_wmma_f32_16x16x4_f32(false, a, false, b2, (short)0, acc2, false, false);
        acc3 = __builtin_amdgcn_wmma_f32_16x16x4_f32(false, a, false, b3, (short)0, acc3, false, false);
    }
}

#if HAVE_ASYNC_LDS
// Stage one 64x32 A tile and 64x32 B tile via async global->LDS (ASYNCcnt).
// Builtin signature (per hipcc diagnostic): first param is int4* (vector ptr).
__device__ __forceinline__ void stage_tile_async(const float* gA, const float* gB,
                                                 float* sA, float* sB, int tid)
{
    #pragma unroll
    for (int i = 0; i < 4; ++i) {
        const int s  = i * 128 + tid;   // 0..511 float4 slots
        const int r  = s >> 3;          // row 0..63
        const int c4 = s & 7;           // float4 column 0..7
        __builtin_amdgcn_global_load_async_to_lds_b128(
            (v4i*)(gA + (size_t)r * H_DIM + c4 * 4),
            (v4i*)(sA + r * LDS_STR + c4 * 4), 0, 0);
        __builtin_amdgcn_global_load_async_to_lds_b128(
            (v4i*)(gB + (size_t)r * H_DIM + c4 * 4),
            (v4i*)(sB + r * LDS_STR + c4 * 4), 0, 0);
    }
}
#endif

// ---------------------------------------------------------------------------
// Kernel 1: router logits GEMM, M=16384 N=64 K=2048, fp32 WMMA 16x16x4,
// double-buffered LDS pipeline (async global->LDS when available).
// ---------------------------------------------------------------------------
__global__ __launch_bounds__(128) void router_gemm_wmma(
    const float* __restrict__ hidden,   // (T, H) row-major
    const float* __restrict__ weight,   // (E, H) row-major
    float* __restrict__ logits)         // (T, E)
{
    __shared__ float ldsA[2][BLOCK_M * LDS_STR];
    __shared__ float ldsB[2][E_EXP   * LDS_STR];

    const int tid   = threadIdx.x;
    const int wave  = tid >> 5;
    const int lane  = tid & 31;
    const int lhalf = lane & 15;          // fragment row (A) / column (B)
    const int khi   = (lane >> 4) << 1;   // lanes 16-31 hold K+2,K+3

    const int row0 = blockIdx.x * BLOCK_M;
    const float* gA = hidden + (size_t)row0 * H_DIM;

    v8f acc0 = {}, acc1 = {}, acc2 = {}, acc3 = {};

#if HAVE_ASYNC_LDS
    // -------- async double-buffered pipeline --------
    stage_tile_async(gA, weight, ldsA[0], ldsB[0], tid);
    wait_async0();
    __syncthreads();

    for (int k0 = 0; k0 < H_DIM; k0 += KB) {
        const int cur = (k0 / KB) & 1;
        if (k0 + KB < H_DIM)  // issue next tile while computing this one
            stage_tile_async(gA + k0 + KB, weight + k0 + KB,
                             ldsA[cur ^ 1], ldsB[cur ^ 1], tid);
        mma_tile(ldsA[cur], ldsB[cur], wave, lhalf, khi, acc0, acc1, acc2, acc3);
        wait_async0();
        __syncthreads();
    }
#else
    // -------- register-staged double-buffered pipeline --------
    float4 ra[4], rb[4];
    #pragma unroll
    for (int i = 0; i < 4; ++i) {
        const int s = i * 128 + tid, r = s >> 3, c4 = s & 7;
        ra[i] = *(const float4*)(gA + (size_t)r * H_DIM + c4 * 4);
        rb[i] = *(const float4*)(weight + (size_t)r * H_DIM + c4 * 4);
    }
    #pragma unroll
    for (int i = 0; i < 4; ++i) {
        const int s = i * 128 + tid, r = s >> 3, c4 = s & 7;
        *(float4*)(&ldsA[0][r * LDS_STR + c4 * 4]) = ra[i];
        *(float4*)(&ldsB[0][r * LDS_STR + c4 * 4]) = rb[i];
    }
    __syncthreads();

    for (int k0 = 0; k0 < H_DIM; k0 += KB) {
        const int cur  = (k0 / KB) & 1;
        const bool more = (k0 + KB) < H_DIM;
        if (more) {
            #pragma unroll
            for (int i = 0; i < 4; ++i) {
                const int s = i * 128 + tid, r = s >> 3, c4 = s & 7;
                ra[i] = *(const float4*)(gA + (size_t)r * H_DIM + k0 + KB + c4 * 4);
                rb[i] = *(const float4*)(weight + (size_t)r * H_DIM + k0 + KB + c4 * 4);
            }
            const int pr = tid >> 1, pc = (tid & 1) * 16;
            __builtin_prefetch(gA + (size_t)pr * H_DIM + k0 + 2 * KB + pc, 0, 3);
            __builtin_prefetch(weight + (size_t)pr * H_DIM + k0 + 2 * KB + pc, 0, 3);
        }
        mma_tile(ldsA[cur], ldsB[cur], wave, lhalf, khi, acc0, acc1, acc2, acc3);
        if (more) {
            #pragma unroll
            for (int i = 0; i < 4; ++i) {
                const int s = i * 128 + tid, r = s >> 3, c4 = s & 7;
                *(float4*)(&ldsA[cur ^ 1][r * LDS_STR + c4 * 4]) = ra[i];
                *(float4*)(&ldsB[cur ^ 1][r * LDS_STR + c4 * 4]) = rb[i];
            }
        }
        __syncthreads();
    }
#endif

    // --- epilogue: C/D layout VGPR r -> row (r + 8*(lane>=16)), lane%16 -> col
    const int mrow = row0 + wave * 16 + (lane >> 4) * 8;
    #pragma unroll
    for (int r = 0; r < 8; ++r) {
        float* p = logits + (size_t)(mrow + r) * E_EXP + lhalf;
        p[ 0] = acc0[r];
        p[16] = acc1[r];
        p[32] = acc2[r];
        p[48] = acc3[r];
    }
}

// ---------------------------------------------------------------------------
// Kernel 2: per-token softmax over 64 experts + top-2 indices over raw logits.
// ---------------------------------------------------------------------------
__global__ __launch_bounds__(256) void router_softmax_topk(
    const float* __restrict__ logits,
    float* __restrict__ affin,
    int* __restrict__ idx)
{
    const int t = blockIdx.x * blockDim.x + threadIdx.x;
    if (t >= T_TOKENS) return;

    const float* row = logits + (size_t)t * E_EXP;
    float v[E_EXP];
    #pragma unroll
    for (int i = 0; i < E_EXP / 4; ++i) {
        const float4 x = *(const float4*)(row + i * 4);
        v[i * 4 + 0] = x.x; v[i * 4 + 1] = x.y;
        v[i * 4 + 2] = x.z; v[i * 4 + 3] = x.w;
    }

    // top-2 with first-occurrence tie-breaking (strict >), like lax.top_k
    float m1 = v[0]; int i1 = 0;
    float m2 = -INFINITY; int i2 = 0;
    #pragma unroll
    for (int j = 1; j < E_EXP; ++j) {
        if (v[j] > m1)      { m2 = m1; i2 = i1; m1 = v[j]; i1 = j; }
        else if (v[j] > m2) { m2 = v[j]; i2 = j; }
    }

    float s = 0.f;
    #pragma unroll
    for (int j = 0; j < E_EXP; ++j) {
        const float e = __expf(v[j] - m1);  // m1 == row max
        v[j] = e;
        s += e;
    }
    const float inv = 1.0f / s;

    float* arow = affin + (size_t)t * E_EXP;
    #pragma unroll
    for (int i = 0; i < E_EXP / 4; ++i) {
        float4 x;
        x.x = v[i * 4 + 0] * inv; x.y = v[i * 4 + 1] * inv;
        x.z = v[i * 4 + 2] * inv; x.w = v[i * 4 + 3] * inv;
        *(float4*)(arow + i * 4) = x;
    }

    idx[t * 2 + 0] = i1;
    idx[t * 2 + 1] = i2;
}

extern "C" void kernel_launch(void* const* d_in, const int* in_sizes, int n_in,
                              void* d_out, int out_size, void* d_ws, size_t ws_size,
                              hipStream_t stream) {
    (void)in_sizes; (void)n_in; (void)d_ws; (void)ws_size; (void)out_size;

    const float* hidden = (const float*)d_in[0];   // (4096,4,2048) -> (16384,2048)
    const float* weight = (const float*)d_in[1];   // (64,2048)
    // d_in[2] = top_k (==2), compile-time constant here

    float* logits = (float*)d_out;                             // T*E
    float* affin  = logits + (size_t)T_TOKENS * E_EXP;         // T*E
    int*   idx    = (int*)(affin + (size_t)T_TOKENS * E_EXP);  // T*2 int32

    router_gemm_wmma<<<T_TOKENS / BLOCK_M, 128, 0, stream>>>(hidden, weight, logits);
    router_softmax_topk<<<T_TOKENS / 256, 256, 0, stream>>>(logits, affin, idx);
}